// DifferentialDropout_v2_66941360276084
// MI455X (gfx1250) — compile-verified
//
#include <hip/hip_runtime.h>
#include <cstdint>
#include <cstddef>

// ---------------------------------------------------------------------------
// DifferentialDropout for x: [64, 256, 56, 56] fp32.  B=64, D=802816.
// Memory-bound (822 MB total traffic ~= 35us @ 23.3 TB/s); Gram matrix uses
// native fp32 WMMA (v_wmma_f32_16x16x4_f32) since compute is ~0.03% of budget
// and fp32 preserves reference numerics.
// ---------------------------------------------------------------------------

typedef __attribute__((ext_vector_type(2))) float v2f;
typedef __attribute__((ext_vector_type(8))) float v8f;

#define BATCH 64
#define NBINS 32   // rounded values clamped to [-16, 15]

// workspace layout (in floats)
#define WS_G     0                  // 64*64 Gram
#define WS_RS    (WS_G + 64 * 64)   // 64 row sums
#define WS_RH    (WS_RS + 64)       // 64*32 row histograms (float counts)
#define WS_P     (WS_RH + 64 * 32)  // 64 keep-drop probabilities p_i
#define WS_SC    (WS_P + 64)        // 1 scale = 1/(1-p[63])
#define WS_TOTAL (WS_SC + 1)

// ---------------------------------------------------------------------------
__global__ void zero_ws_kernel(float* __restrict__ ws, int n) {
  int i = blockIdx.x * blockDim.x + threadIdx.x;
  if (i < n) ws[i] = 0.0f;
}

// ---------------------------------------------------------------------------
// Pass 1: per-row sum + 32-bin histogram of rounded values.
// grid = (chunks_per_row, 64 rows), block = 256.
__global__ void stats_kernel(const float* __restrict__ x,
                             float* __restrict__ row_sum,
                             float* __restrict__ row_hist,
                             int D, int cpb) {
  __shared__ float hist[NBINS];
  __shared__ float red[256];

  const int row  = blockIdx.y;
  const int base = blockIdx.x * cpb;
  int end = base + cpb;
  if (end > D) end = D;

  for (int i = threadIdx.x; i < NBINS; i += blockDim.x) hist[i] = 0.0f;
  __syncthreads();

  const float* __restrict__ xr = x + (size_t)row * (size_t)D;
  float s = 0.0f;
  for (int i = base + threadIdx.x; i < end; i += blockDim.x) {
    float v = xr[i];
    s += v;
    int b = (int)rintf(v) + 16;        // rintf == round-half-even == jnp.round
    b = (b < 0) ? 0 : ((b > 31) ? 31 : b);
    atomicAdd(&hist[b], 1.0f);         // ds_add_f32
  }
  red[threadIdx.x] = s;
  __syncthreads();
  for (int off = blockDim.x >> 1; off > 0; off >>= 1) {
    if ((int)threadIdx.x < off) red[threadIdx.x] += red[threadIdx.x + off];
    __syncthreads();
  }
  if (threadIdx.x == 0) atomicAdd(&row_sum[row], red[0]);
  if ((int)threadIdx.x < NBINS)
    atomicAdd(&row_hist[row * NBINS + threadIdx.x], hist[threadIdx.x]);
}

// ---------------------------------------------------------------------------
// Pass 2: G = temp @ temp^T via V_WMMA_F32_16X16X4_F32.
// block = 512 threads = 16 waves; wave w computes 16x16 tile (w>>2, w&3).
// A fragment (16x4, f32): lane l holds row M=l%16, K pair (l/16)*2..+1 in 2 VGPRs.
// B fragment (4x16): lane l holds col N=l%16 of the column block with the same
// K-pair striping, i.e. identical load shape from the other row block.
// blockIdx.x splits K in chunks of `kc`; one atomic flush per wave at the end.
__global__ void gram_wmma_kernel(const float* __restrict__ x,
                                 float* __restrict__ G,
                                 int D, int kc) {
  const int wave  = threadIdx.x >> 5;
  const int lane  = threadIdx.x & 31;
  const int ti    = wave >> 2;        // 0..3 row-block
  const int tj    = wave & 3;         // 0..3 col-block
  const int m     = lane & 15;
  const int khalf = (lane >> 4) << 1; // 0 or 2

  const float* __restrict__ arow =
      x + (size_t)(16 * ti + m) * (size_t)D + (size_t)khalf;
  const float* __restrict__ brow =
      x + (size_t)(16 * tj + m) * (size_t)D + (size_t)khalf;

  int k0 = blockIdx.x * kc;
  int k1 = k0 + kc;
  if (k1 > D) k1 = D;

  v8f acc = {0.f, 0.f, 0.f, 0.f, 0.f, 0.f, 0.f, 0.f};

#pragma unroll 4
  for (int k = k0; k < k1; k += 4) {
    v2f a = *(const v2f*)(arow + k);   // global_load_b64
    v2f b = *(const v2f*)(brow + k);
    acc = __builtin_amdgcn_wmma_f32_16x16x4_f32(
        /*neg_a=*/false, a, /*neg_b=*/false, b,
        /*c_mod=*/(short)0, acc, /*reuse_a=*/false, /*reuse_b=*/false);
  }

  // C/D layout: VGPR v, lanes 0-15 -> M=v ; lanes 16-31 -> M=v+8 ; N = lane%16
  const int hi = lane >> 4;
#pragma unroll
  for (int v = 0; v < 8; ++v) {
    int gm = 16 * ti + v + hi * 8;
    int gn = 16 * tj + m;
    atomicAdd(&G[gm * BATCH + gn], acc[v]);   // global_atomic_add_f32
  }
}

// ---------------------------------------------------------------------------
// Finalize: everything on the 64x64 Gram + histograms. 1 block, 64 threads.
// cov_ij = G_ij - s_i s_j / D   (centering folds into rank-1 correction)
// row_mse_i = (G_ii - (2/B) sum_j G_ij + (1/B^2) sum_jk G_jk) / D
__global__ void finalize_kernel(float* __restrict__ ws, int D) {
  float* __restrict__ G  = ws + WS_G;
  float* __restrict__ rs = ws + WS_RS;
  float* __restrict__ rh = ws + WS_RH;
  float* __restrict__ p  = ws + WS_P;
  float* __restrict__ sc = ws + WS_SC;

  __shared__ float covd[64], rGs[64], rmse[64], bh[NBINS], sh[4];

  const int i = threadIdx.x;           // 0..63
  const float invD = 1.0f / (float)D;
  const float si = rs[i];
  const float Gii = G[i * BATCH + i];
  covd[i] = Gii - si * si * invD;
  __syncthreads();

  float absmean = 0.0f, gsum = 0.0f;
  for (int j = 0; j < BATCH; ++j) {
    float gij = G[i * BATCH + j];
    gsum += gij;
    float covij = gij - si * rs[j] * invD;
    float c = covij * rsqrtf(covd[i] * covd[j]);
    absmean += fabsf(c);
  }
  absmean *= (1.0f / (float)BATCH);
  rGs[i] = gsum;
  __syncthreads();

  if (i == 0) {
    float t = 0.0f;
    for (int j = 0; j < BATCH; ++j) t += rGs[j];
    sh[0] = t;
  }
  __syncthreads();
  const float totG = sh[0];
  const float mse =
      (Gii - (2.0f / (float)BATCH) * gsum + totG / ((float)BATCH * (float)BATCH)) * invD;
  rmse[i] = mse;
  __syncthreads();

  if (i == 0) {
    float t = 0.0f;
    for (int j = 0; j < BATCH; ++j) t += rmse[j];
    sh[1] = t;
  }
  if (i < NBINS) {               // batch histogram = sum of row histograms
    float t = 0.0f;
    for (int r = 0; r < BATCH; ++r) t += rh[r * NBINS + i];
    bh[i] = t;
  }
  __syncthreads();
  if (i == 0) {                  // batch entropy
    float N = (float)BATCH * (float)D, H = 0.0f;
    for (int b = 0; b < NBINS; ++b) {
      float c = bh[b];
      if (c > 0.0f) { float q = c / N; H -= q * log2f(q); }
    }
    sh[2] = H;
  }
  __syncthreads();
  const float totmse = sh[1];
  const float bent   = sh[2];

  float Hr = 0.0f;               // row entropy
  const float Nf = (float)D;
  for (int b = 0; b < NBINS; ++b) {
    float c = rh[i * NBINS + b];
    if (c > 0.0f) { float q = c / Nf; Hr -= q * log2f(q); }
  }

  const float f2    = mse / totmse;
  const float ratio = Hr / bent;
  const float f3    = fminf(ratio, 1.0f / ratio);
  const float pi    = (1.0f - absmean) * f2 * f3;
  p[i] = pi;
  if (i == BATCH - 1) sc[0] = 1.0f / (1.0f - pi);   // reference uses p[last]
}

// ---------------------------------------------------------------------------
// Pass 3: out = (u > p_row) ? x * scale : 0.  Deterministic hash RNG
// (JAX threefry stream is not bit-reproducible here; no runtime check).
__device__ __forceinline__ float u01_hash(uint32_t v) {
  v ^= v >> 17; v *= 0xed5ad4bbu;
  v ^= v >> 11; v *= 0xac4c1b51u;
  v ^= v >> 15; v *= 0x31848babu;
  v ^= v >> 14;
  return (float)(v >> 8) * (1.0f / 16777216.0f);
}

__global__ void apply_kernel(const float* __restrict__ x,
                             const float* __restrict__ p,
                             const float* __restrict__ scale,
                             float* __restrict__ out,
                             int D, size_t total4) {
  const float sc = scale[0];
  const size_t stride = (size_t)gridDim.x * blockDim.x;
  for (size_t i = (size_t)blockIdx.x * blockDim.x + threadIdx.x; i < total4;
       i += stride) {
    size_t e = i * 4;                      // D % 4 == 0 -> row uniform in a float4
    int row = (int)(e / (size_t)D);
    float pr = p[row];
    float4 v = reinterpret_cast<const float4*>(x)[i];
    float4 o;
    o.x = (u01_hash((uint32_t)e)     > pr) ? v.x * sc : 0.0f;
    o.y = (u01_hash((uint32_t)e + 1) > pr) ? v.y * sc : 0.0f;
    o.z = (u01_hash((uint32_t)e + 2) > pr) ? v.z * sc : 0.0f;
    o.w = (u01_hash((uint32_t)e + 3) > pr) ? v.w * sc : 0.0f;
    reinterpret_cast<float4*>(out)[i] = o;
  }
}

// ---------------------------------------------------------------------------
extern "C" void kernel_launch(void* const* d_in, const int* in_sizes, int n_in,
                              void* d_out, int out_size, void* d_ws, size_t ws_size,
                              hipStream_t stream) {
  const float* x = (const float*)d_in[0];   // [64, 256, 56, 56] fp32
  float* out = (float*)d_out;
  float* ws  = (float*)d_ws;

  const int total = in_sizes[0];            // 51,380,224
  const int D = total / BATCH;              // 802,816

  // 0) zero workspace accumulators (harness does not re-poison between replays)
  zero_ws_kernel<<<(WS_TOTAL + 255) / 256, 256, 0, stream>>>(ws, WS_TOTAL);

  // 1) row sums + row histograms (one full read of x)
  const int chunks = 64;
  const int cpb = (D + chunks - 1) / chunks;
  stats_kernel<<<dim3(chunks, BATCH), 256, 0, stream>>>(
      x, ws + WS_RS, ws + WS_RH, D, cpb);

  // 2) Gram via fp32 WMMA (second full read of x)
  const int kc = 8192;                       // 802816 / 8192 = 98 blocks exactly
  const int nb = (D + kc - 1) / kc;
  gram_wmma_kernel<<<nb, 512, 0, stream>>>(x, ws + WS_G, D, kc);

  // 3) tiny finalize: p[64] + scale
  finalize_kernel<<<1, 64, 0, stream>>>(ws, D);

  // 4) masked rescale (third read + one write)
  const size_t total4 = (size_t)total / 4;
  apply_kernel<<<2048, 256, 0, stream>>>(x, ws + WS_P, ws + WS_SC, out, D, total4);
}